// MultiHeadAttentionQuantum_65481071406742
// MI455X (gfx1250) — compile-verified
//
#include <hip/hip_runtime.h>
#include <math.h>

typedef __attribute__((ext_vector_type(2))) float v2f;
typedef __attribute__((ext_vector_type(8))) float v8f;

#define NHEADS   16
#define DK       8
#define SEQ      2048
#define EMB      128   // NHEADS*DK
#define BATCH    4
#define KV_PITCH 9     // padded LDS row stride (floats) to spread banks

// scores are statically bounded: |q.k|<=8 (cos in [-1,1], dk=8).
// Q fragments pre-scaled by (1/sqrt(8))*log2(e); shift = 8*that bound.
#define QSCALE   (0.35355339059327373f * 1.4426950408889634f)
#define SHIFT2   (8.0f * 0.35355339059327373f * 1.4426950408889634f)

// ---------------------------------------------------------------------------
// Kernel 1: qkv = cos(x + theta), transposed to [B*H][S][DK] for attention
// ---------------------------------------------------------------------------
__global__ __launch_bounds__(256)
void qcos_kernel(const float* __restrict__ x, const float* __restrict__ theta,
                 float* __restrict__ q) {
    int idx = blockIdx.x * 256 + threadIdx.x;      // [0, B*S*E)
    if (idx >= BATCH * SEQ * EMB) return;
    int d = idx & 7;
    int h = (idx >> 3) & 15;
    int s = (idx >> 7) & 2047;
    int b = idx >> 18;
    float v = __cosf(x[idx] + theta[d]);
    q[(((size_t)(b * NHEADS + h) * SEQ) + s) * DK + d] = v;
}

// ---------------------------------------------------------------------------
// Kernel 2: streaming attention, q=k=v, per (b,h), fixed-shift softmax.
// grid.x = B*H*16 ; block = 256 (8 waves); each wave owns 16 queries.
// dynamic LDS: SEQ*KV_PITCH floats (73728 B)
// ---------------------------------------------------------------------------
__global__ __launch_bounds__(256)
void attn_kernel(const float* __restrict__ qg, float* __restrict__ og) {
    extern __shared__ float kv[];          // [SEQ][KV_PITCH]

    const int tid  = threadIdx.x;
    const int bh   = blockIdx.x >> 4;      // (b*H + h)
    const int qblk = blockIdx.x & 15;

    // ---- stage K (=Q=V) for this (b,h) into LDS with padded pitch ----
    const float4* src4 = reinterpret_cast<const float4*>(qg + (size_t)bh * SEQ * DK);
    for (int i = tid; i < SEQ * DK / 4; i += 256) {
        float4 v = src4[i];
        int srow = i >> 1;                 // 2 float4 per 8-float row
        int dcol = (i & 1) * 4;
        float* dst = &kv[srow * KV_PITCH + dcol];
        dst[0] = v.x; dst[1] = v.y; dst[2] = v.z; dst[3] = v.w;
    }
    __syncthreads();

    const int  wave  = tid >> 5;
    const int  lane  = tid & 31;
    const int  ln    = lane & 15;
    const bool hi    = (lane >= 16);
    const int  off   = hi ? 2 : 0;
    const int  qrow  = qblk * 128 + wave * 16 + ln;   // this lane's query

    // ---- Q^T B-fragments (fixed for the whole key loop), pre-scaled ----
    v2f bq0, bq1;
    bq0.x = kv[qrow * KV_PITCH + off + 0] * QSCALE;
    bq0.y = kv[qrow * KV_PITCH + off + 1] * QSCALE;
    bq1.x = kv[qrow * KV_PITCH + 4 + off + 0] * QSCALE;
    bq1.y = kv[qrow * KV_PITCH + 4 + off + 1] * QSCALE;

    v8f oacc = {0.f, 0.f, 0.f, 0.f, 0.f, 0.f, 0.f, 0.f};
    float l = 0.f;   // per-lane partial denominator (this lane's 8-key slices)

    for (int kt = 0; kt < SEQ / 16; ++kt) {
        const int kbase = kt * 16;
        const int krow  = kbase + ln;

        // A = K tile [16k x d] fragments
        v2f a0, a1;
        a0.x = kv[krow * KV_PITCH + off + 0];
        a0.y = kv[krow * KV_PITCH + off + 1];
        a1.x = kv[krow * KV_PITCH + 4 + off + 0];
        a1.y = kv[krow * KV_PITCH + 4 + off + 1];

        // S^T[16k x 16q] = K . Q^T  (two K=4 steps over dk=8), log2-domain
        v8f s8 = {0.f, 0.f, 0.f, 0.f, 0.f, 0.f, 0.f, 0.f};
        s8 = __builtin_amdgcn_wmma_f32_16x16x4_f32(false, a0, false, bq0,
                                                   (short)0, s8, false, false);
        s8 = __builtin_amdgcn_wmma_f32_16x16x4_f32(false, a1, false, bq1,
                                                   (short)0, s8, false, false);

        // ---- fixed-shift softmax numerators: p = 2^(s - SHIFT2).
        // argument is in [-8.16, 0]: raw v_exp_f32 is exact-safe here, so
        // bypass libm's denormal-range guard with the raw builtin.
        float p[8];
        float tsum = 0.f;
        #pragma unroll
        for (int r = 0; r < 8; ++r) {
            p[r] = __builtin_amdgcn_exp2f(s8[r] - SHIFT2);
            tsum += p[r];
        }
        l += tsum;                          // cross-half combine deferred to end

        // ---- build P^T B-fragments: 4 half-dependent shuffles ----
        float t1 = __shfl_xor(hi ? p[0] : p[2], 16, 32);
        float t2 = __shfl_xor(hi ? p[1] : p[3], 16, 32);
        float t3 = __shfl_xor(hi ? p[4] : p[6], 16, 32);
        float t4 = __shfl_xor(hi ? p[5] : p[7], 16, 32);

        v2f pb0, pb1, pb2, pb3;
        pb0.x = hi ? t1   : p[0];  pb0.y = hi ? t2   : p[1];   // keys 0..3
        pb1.x = hi ? t3   : p[4];  pb1.y = hi ? t4   : p[5];   // keys 4..7
        pb2.x = hi ? p[2] : t1;    pb2.y = hi ? p[3] : t2;     // keys 8..11
        pb3.x = hi ? p[6] : t3;    pb3.y = hi ? p[7] : t4;     // keys 12..15

        // ---- A = V^T chunks; O^T[d x 16q] += V^T . P^T ----
        v2f av;
        av.x = kv[(kbase + 0 + off) * KV_PITCH + ln];
        av.y = kv[(kbase + 1 + off) * KV_PITCH + ln];
        oacc = __builtin_amdgcn_wmma_f32_16x16x4_f32(false, av, false, pb0,
                                                     (short)0, oacc, false, false);
        av.x = kv[(kbase + 4 + off) * KV_PITCH + ln];
        av.y = kv[(kbase + 5 + off) * KV_PITCH + ln];
        oacc = __builtin_amdgcn_wmma_f32_16x16x4_f32(false, av, false, pb1,
                                                     (short)0, oacc, false, false);
        av.x = kv[(kbase + 8 + off) * KV_PITCH + ln];
        av.y = kv[(kbase + 9 + off) * KV_PITCH + ln];
        oacc = __builtin_amdgcn_wmma_f32_16x16x4_f32(false, av, false, pb2,
                                                     (short)0, oacc, false, false);
        av.x = kv[(kbase + 12 + off) * KV_PITCH + ln];
        av.y = kv[(kbase + 13 + off) * KV_PITCH + ln];
        oacc = __builtin_amdgcn_wmma_f32_16x16x4_f32(false, av, false, pb3,
                                                     (short)0, oacc, false, false);
    }

    // ---- single denominator reduction + normalize + store ----
    float invl = 1.f / (l + __shfl_xor(l, 16, 32));
    if (!hi) {
        const int b = bh >> 4;
        const int h = bh & 15;
        float4 r0, r1;
        r0.x = oacc[0] * invl; r0.y = oacc[1] * invl;
        r0.z = oacc[2] * invl; r0.w = oacc[3] * invl;
        r1.x = oacc[4] * invl; r1.y = oacc[5] * invl;
        r1.z = oacc[6] * invl; r1.w = oacc[7] * invl;
        float* dst = og + ((size_t)(b * SEQ + qrow)) * EMB + h * DK;
        reinterpret_cast<float4*>(dst)[0] = r0;
        reinterpret_cast<float4*>(dst)[1] = r1;
    }
}

// ---------------------------------------------------------------------------
// Kernel 3: out[8192x128] = O @ W^T + b, fp32 WMMA, each wave: 16 rows x 128
// grid.x = 128, block = 128 (4 waves, 64 rows/block)
// ---------------------------------------------------------------------------
__global__ __launch_bounds__(128)
void proj_kernel(const float* __restrict__ O2, const float* __restrict__ W,
                 const float* __restrict__ bias, float* __restrict__ out) {
    const int  wave = threadIdx.x >> 5;
    const int  lane = threadIdx.x & 31;
    const int  ln   = lane & 15;
    const bool hi   = (lane >= 16);
    const int  off  = hi ? 2 : 0;
    const int  rowbase = (blockIdx.x * 4 + wave) * 16;

    v8f acc[8];
    #pragma unroll
    for (int n = 0; n < 8; ++n) acc[n] = (v8f){0.f,0.f,0.f,0.f,0.f,0.f,0.f,0.f};

    for (int c = 0; c < 32; ++c) {
        const int kd = 4 * c + off;
        const float* ar = O2 + (size_t)(rowbase + ln) * EMB + kd;
        v2f a; a.x = ar[0]; a.y = ar[1];
        #pragma unroll
        for (int n = 0; n < 8; ++n) {
            const float* wr = W + (size_t)(n * 16 + ln) * EMB + kd;  // W^T fragment
            v2f bw; bw.x = wr[0]; bw.y = wr[1];
            acc[n] = __builtin_amdgcn_wmma_f32_16x16x4_f32(false, a, false, bw,
                                                           (short)0, acc[n],
                                                           false, false);
        }
    }

    #pragma unroll
    for (int n = 0; n < 8; ++n) {
        const int col = n * 16 + ln;
        const float bb = bias[col];
        #pragma unroll
        for (int r = 0; r < 8; ++r) {
            const int row = rowbase + r + (hi ? 8 : 0);
            out[(size_t)row * EMB + col] = acc[n][r] + bb;
        }
    }
}

// ---------------------------------------------------------------------------
extern "C" void kernel_launch(void* const* d_in, const int* in_sizes, int n_in,
                              void* d_out, int out_size, void* d_ws, size_t ws_size,
                              hipStream_t stream) {
    const float* x     = (const float*)d_in[0];   // [4,2048,128]
    const float* theta = (const float*)d_in[1];   // [8]
    const float* W     = (const float*)d_in[2];   // [128,128]
    const float* bias  = (const float*)d_in[3];   // [128]
    float* out = (float*)d_out;                   // [4,2048,128]

    float* ws1 = (float*)d_ws;                       // [B*H][S][DK]  (4 MB)
    float* ws2 = ws1 + (size_t)BATCH * SEQ * EMB;    // [B*S][E]      (4 MB)

    const int n_elem = BATCH * SEQ * EMB;
    qcos_kernel<<<(n_elem + 255) / 256, 256, 0, stream>>>(x, theta, ws1);

    const size_t lds_bytes = (size_t)SEQ * KV_PITCH * sizeof(float);  // 73728
    attn_kernel<<<BATCH * NHEADS * 16, 256, lds_bytes, stream>>>(ws1, ws2);

    proj_kernel<<<128, 128, 0, stream>>>(ws2, W, bias, out);
}